// LightGCN_9491877724638
// MI455X (gfx1250) — compile-verified
//
#include <hip/hip_runtime.h>
#include <cstdint>

// LightGCN, 2 layers, D=64, f32 on MI455X (gfx1250, wave32).
// Strategy: build CSR once (deg -> exclusive scan -> slot fill, 4M u32 atomics
// total), then each layer is a pure gather-reduce: one wave per node, lane l
// owns dims (2l,2l+1). 38.4MB node table is L2-resident (192MB L2); each wave
// stages its 32-edge CSR chunk into wave-private LDS with
// global_load_async_to_lds_b32 + s_wait_asynccnt (ASYNCcnt is per-wave, so no
// block barrier). Zero f32 atomics; every output row stored exactly once.

#define SCAN_B 1024

__device__ __forceinline__ uint32_t lds_off(const void* p) {
  // addrspace(3) pointers carry the workgroup-relative LDS byte offset in the
  // low 32 bits of the generic pointer.
  return (uint32_t)(uintptr_t)p;
}

__device__ __forceinline__ void async_load_dword_to_lds(uint32_t lds_byte_addr,
                                                        const void* base,
                                                        uint32_t byte_off) {
  // GVS mode: copies a dword from the global base in saddr plus the offset in vaddr to the LDS address in vdst; tracked with ASYNCcnt.
  asm volatile("global_load_async_to_lds_b32 %0, %1, %2"
               :
               : "v"(lds_byte_addr), "v"(byte_off),
                 "s"((uint64_t)(uintptr_t)base)
               : "memory");
}

__device__ __forceinline__ void wait_async0() {
  asm volatile("s_wait_asynccnt 0" ::: "memory");
}

__global__ __launch_bounds__(256) void k_zero_u32(uint32_t* __restrict__ p,
                                                  long long n) {
  long long i = (long long)blockIdx.x * blockDim.x + threadIdx.x;
  if (i < n) p[i] = 0u;
}

__global__ __launch_bounds__(256) void k_deg(const long long* __restrict__ row,
                                             uint32_t* __restrict__ deg,
                                             long long E) {
  long long e = (long long)blockIdx.x * blockDim.x + threadIdx.x;
  if (e < E) {
    atomicAdd(&deg[(uint32_t)row[e]], 1u);
  }
}

__global__ __launch_bounds__(256) void k_dinv(const uint32_t* __restrict__ deg,
                                              float* __restrict__ dinv,
                                              long long N) {
  long long i = (long long)blockIdx.x * blockDim.x + threadIdx.x;
  if (i < N) {
    uint32_t d = deg[i];
    dinv[i] = d ? rsqrtf((float)d) : 0.0f;
  }
}

// Exclusive scan, level 1: per-block Hillis-Steele over 1024 elements.
__global__ __launch_bounds__(SCAN_B) void k_scan1(
    const uint32_t* __restrict__ deg, uint32_t* __restrict__ rowStart,
    uint32_t* __restrict__ blockSums, long long N) {
  __shared__ uint32_t s[SCAN_B];
  const int tid = threadIdx.x;
  const long long i = (long long)blockIdx.x * SCAN_B + tid;
  const uint32_t x = (i < N) ? deg[i] : 0u;
  s[tid] = x;
  __syncthreads();
  for (int o = 1; o < SCAN_B; o <<= 1) {
    uint32_t t = (tid >= o) ? s[tid - o] : 0u;
    __syncthreads();
    s[tid] += t;
    __syncthreads();
  }
  if (i < N) rowStart[i] = s[tid] - x;  // exclusive within block
  if (tid == SCAN_B - 1) blockSums[blockIdx.x] = s[tid];
}

// Exclusive scan, level 2: single block over <=1024 block sums (N <= 1M).
__global__ __launch_bounds__(SCAN_B) void k_scan2(
    uint32_t* __restrict__ blockSums, int nb) {
  __shared__ uint32_t s[SCAN_B];
  const int tid = threadIdx.x;
  const uint32_t x = (tid < nb) ? blockSums[tid] : 0u;
  s[tid] = x;
  __syncthreads();
  for (int o = 1; o < SCAN_B; o <<= 1) {
    uint32_t t = (tid >= o) ? s[tid - o] : 0u;
    __syncthreads();
    s[tid] += t;
    __syncthreads();
  }
  if (tid < nb) blockSums[tid] = s[tid] - x;  // exclusive
}

__global__ __launch_bounds__(256) void k_scan3(
    uint32_t* __restrict__ rowStart, const uint32_t* __restrict__ blockSums,
    long long N, uint32_t Etot) {
  long long i = (long long)blockIdx.x * blockDim.x + threadIdx.x;
  if (i < N) {
    rowStart[i] += blockSums[i / SCAN_B];
  } else if (i == N) {
    rowStart[N] = Etot;
  }
}

__global__ __launch_bounds__(256) void k_fill(
    const long long* __restrict__ eidx, const uint32_t* __restrict__ rowStart,
    uint32_t* __restrict__ cursor, uint32_t* __restrict__ csrCol,
    long long E) {
  long long e = (long long)blockIdx.x * blockDim.x + threadIdx.x;
  if (e < E) {
    const uint32_t r = (uint32_t)eidx[e];
    const uint32_t c = (uint32_t)eidx[E + e];
    const uint32_t slot = atomicAdd(&cursor[r], 1u);
    csrCol[rowStart[r] + slot] = c;
  }
}

// One wave per node. dst[n] = src[n] + sum_{c in neigh(n)} dinv[n]*dinv[c]*src[c].
__global__ __launch_bounds__(256) void k_gather(
    const uint32_t* __restrict__ rowStart, const uint32_t* __restrict__ csrCol,
    const float* __restrict__ dinv, const float* __restrict__ src,
    float* __restrict__ dst, long long N) {
  __shared__ uint32_t s_col[256];  // 32 entries per wave (wave-private)
  __shared__ float s_nrm[256];
  const int w = threadIdx.x >> 5;
  const int lane = threadIdx.x & 31;
  const long long n = (long long)blockIdx.x * 8 + w;
  if (n >= N) return;  // wave-uniform; no block barriers in this kernel

  const uint32_t beg = rowStart[n];
  const uint32_t end = rowStart[n + 1];
  const float dr = dinv[n];
  uint32_t* mycol = &s_col[w * 32];
  float* mynrm = &s_nrm[w * 32];

  float ax = 0.0f, ay = 0.0f;
  for (uint32_t j0 = beg; j0 < end; j0 += 32u) {
    uint32_t j = j0 + (uint32_t)lane;
    if (j >= end) j = end - 1u;  // clamp: all 32 lanes issue valid loads
    async_load_dword_to_lds(lds_off(&mycol[lane]), csrCol, j << 2);
    wait_async0();  // per-wave ASYNCcnt; chunk is wave-private
    // one coalesced dinv gather per chunk instead of 32 broadcast loads
    mynrm[lane] = dr * dinv[mycol[lane]];
    const uint32_t rem = end - j0;
    const int len = (int)(rem < 32u ? rem : 32u);
    for (int k = 0; k < len; ++k) {
      const uint32_t c = mycol[k];     // LDS broadcast (uniform)
      const float nrm = mynrm[k];      // LDS broadcast (uniform)
      const float2 v = ((const float2*)(src + (size_t)c * 64))[lane];
      ax = fmaf(nrm, v.x, ax);
      ay = fmaf(nrm, v.y, ay);
    }
  }
  const float2 self = ((const float2*)(src + (size_t)n * 64))[lane];
  float2 o;
  o.x = self.x + ax;
  o.y = self.y + ay;
  ((float2*)(dst + (size_t)n * 64))[lane] = o;
}

extern "C" void kernel_launch(void* const* d_in, const int* in_sizes, int n_in,
                              void* d_out, int out_size, void* d_ws,
                              size_t ws_size, hipStream_t stream) {
  (void)n_in;
  (void)out_size;
  (void)ws_size;

  const float* user = (const float*)d_in[0];
  const float* item = (const float*)d_in[1];
  const long long* eidx = (const long long*)d_in[2];

  const long long U = (long long)in_sizes[0] / 64;
  const long long I = (long long)in_sizes[1] / 64;
  const long long N = U + I;
  const long long E = (long long)in_sizes[2] / 2;

  // workspace layout (256B-aligned slots)
  char* ws = (char*)d_ws;
  auto align256 = [](size_t x) { return (x + 255) & ~(size_t)255; };
  size_t off = 0;
  uint32_t* deg = (uint32_t*)(ws + off);       off += align256((size_t)N * 4);
  uint32_t* cursor = (uint32_t*)(ws + off);    off += align256((size_t)N * 4);
  float* dinv = (float*)(ws + off);            off += align256((size_t)N * 4);
  uint32_t* rowStart = (uint32_t*)(ws + off);  off += align256(((size_t)N + 1) * 4);
  uint32_t* blockSums = (uint32_t*)(ws + off); off += align256((size_t)SCAN_B * 4);
  uint32_t* csrCol = (uint32_t*)(ws + off);    off += align256((size_t)E * 4);
  float* tmp = (float*)(ws + off);             // [N,64] f32

  float* out = (float*)d_out;

  const unsigned nBlk = (unsigned)((N + 255) / 256);
  const unsigned eBlk = (unsigned)((E + 255) / 256);
  const int nb1 = (int)((N + SCAN_B - 1) / SCAN_B);

  // all_emb^(0) = concat(user, item) directly into d_out
  hipMemcpyAsync(out, user, (size_t)U * 64 * sizeof(float),
                 hipMemcpyDeviceToDevice, stream);
  hipMemcpyAsync(out + (size_t)U * 64, item, (size_t)I * 64 * sizeof(float),
                 hipMemcpyDeviceToDevice, stream);

  // degree + rsqrt (edge list constant -> identical for both layers)
  k_zero_u32<<<nBlk, 256, 0, stream>>>(deg, N);
  k_zero_u32<<<nBlk, 256, 0, stream>>>(cursor, N);
  k_deg<<<eBlk, 256, 0, stream>>>(eidx, deg, E);
  k_dinv<<<nBlk, 256, 0, stream>>>(deg, dinv, N);

  // CSR: rowStart = exclusive_scan(deg); csrCol = cols grouped by row
  k_scan1<<<(unsigned)nb1, SCAN_B, 0, stream>>>(deg, rowStart, blockSums, N);
  k_scan2<<<1, SCAN_B, 0, stream>>>(blockSums, nb1);
  k_scan3<<<(unsigned)((N + 1 + 255) / 256), 256, 0, stream>>>(
      rowStart, blockSums, N, (uint32_t)E);
  k_fill<<<eBlk, 256, 0, stream>>>(eidx, rowStart, cursor, csrCol, E);

  const unsigned gBlk = (unsigned)((N + 7) / 8);  // 8 waves (nodes) per block

  // Layer 1: tmp = out + A_norm * out
  k_gather<<<gBlk, 256, 0, stream>>>(rowStart, csrCol, dinv, out, tmp, N);
  // Layer 2: out = tmp + A_norm * tmp   (final result lands in d_out)
  k_gather<<<gBlk, 256, 0, stream>>>(rowStart, csrCol, dinv, tmp, out, N);
}